// Sparsemax_39341900431900
// MI455X (gfx1250) — compile-verified
//
#include <hip/hip_runtime.h>
#include <hip/hip_bf16.h>
#include <cstdint>

// Sparsemax over last dim, rows of D=8192 fp32 (4096 rows).
// Memory-bound: 256MB total traffic -> ~11us floor at 23.3 TB/s.
// Data path: Tensor Data Mover (TDM) DMA global->LDS, one descriptor per row,
// double-buffered and tracked with TENSORcnt; register-resident Newton solve
// (exact for the piecewise-linear sparsemax threshold equation) with wave32
// shuffle + single-barrier ping-pong block reductions; NT 128-bit stores out.

#define D 8192
#define T 256                 // threads per block (8 wave32)
#define NWAVES (T / 32)
#define CPT (D / 4 / T)       // float4 chunks per thread = 8
#define NEWTON_MAX 24         // uniform early-exit typically fires after ~6-8

typedef float v4f  __attribute__((ext_vector_type(4)));
typedef unsigned int u32x4 __attribute__((ext_vector_type(4)));
typedef unsigned int u32x8 __attribute__((ext_vector_type(8)));

static_assert(CPT == 8, "layout assumption");

__device__ __forceinline__ float wave_rmax(float v) {
#pragma unroll
  for (int o = 16; o > 0; o >>= 1) v = fmaxf(v, __shfl_xor(v, o, 32));
  return v;
}

__device__ __forceinline__ float wave_rsum(float v) {
#pragma unroll
  for (int o = 16; o > 0; o >>= 1) v += __shfl_xor(v, o, 32);
  return v;
}

__global__ __launch_bounds__(T) void sparsemax_rows_kernel(const float* __restrict__ x,
                                                           float* __restrict__ out,
                                                           int rows) {
  __shared__ float buf[2][D];             // 64KB double buffer (320KB LDS per WGP)
  __shared__ float red_s[2][NWAVES];      // ping-pong reduction banks
  __shared__ float red_k[2][NWAVES];

  const int t = threadIdx.x;
  const int lane = t & 31;
  const int w = t >> 5;
  const int stride = gridDim.x;

  const uint32_t lds_base[2] = {(uint32_t)(uintptr_t)(&buf[0][0]),
                                (uint32_t)(uintptr_t)(&buf[1][0])};

  // One TDM descriptor DMA per row: 2D tensor/tile 8192x1, 4B elements.
  // Group1 is row-invariant (hoisted by the compiler); group0 carries the
  // per-row global address and per-buffer LDS address.
  auto issue_tdm = [&](int row, int b) {
    const uint64_t ga = (uint64_t)(uintptr_t)(x + (size_t)row * D);

    u32x4 g0;
    g0.x = 1u;                                   // count=1 (valid), user mode
    g0.y = lds_base[b];                          // lds_addr (bytes)
    g0.z = (uint32_t)ga;                         // global_addr[31:0]
    g0.w = (uint32_t)(ga >> 32) | 0x80000000u;   // global_addr[56:32] | type=2<<30

    u32x8 g1;
    g1.s0 = 2u << 16;                            // wg_mask=0 | data_size=4B
    g1.s1 = (uint32_t)(D & 0xFFFF) << 16;        // tensor_dim0[15:0] = 8192
    g1.s2 = 1u << 16;                            // tensor_dim0 hi=0 | tensor_dim1=1
    g1.s3 = (uint32_t)(D & 0xFFFF) << 16;        // tensor_dim1 hi=0 | tile_dim0=8192
    g1.s4 = 1u;                                  // tile_dim1=1 | tile_dim2=0
    g1.s5 = (uint32_t)D;                         // tensor_dim0_stride = 8192
    g1.s6 = 0u;                                  // stride hi=0 | dim1_stride lo=0
    g1.s7 = 0u;                                  // dim1_stride hi=0

    asm volatile("tensor_load_to_lds %0, %1"
                 :
                 : "s"(g0), "s"(g1)
                 : "memory");
  };

  const int row0 = blockIdx.x;
  if (w == 0 && row0 < rows) issue_tdm(row0, 0);  // prologue DMA (wave 0 only)

  for (int i = 0, row = row0; row < rows; ++i, row += stride) {
    const int b = i & 1;
    const int nrow = row + stride;
    if (w == 0) {
      if (nrow < rows) {
        // Overwrites buf[b^1]: all waves finished reading it before the
        // trailing barrier of the previous iteration, so this is safe.
        issue_tdm(nrow, b ^ 1);
        // TDM ops from one wave complete in order: cnt<=1 => current row done.
        asm volatile("s_wait_tensorcnt 0x1" ::: "memory");
      } else {
        asm volatile("s_wait_tensorcnt 0x0" ::: "memory");
      }
    }
    __syncthreads();            // row data visible to all 8 waves

    // Pull this row into registers (32 floats/thread) via LDS.
    v4f v[CPT];
    const v4f* b4 = (const v4f*)(&buf[b][0]);
#pragma unroll
    for (int j = 0; j < CPT; ++j) v[j] = b4[t + j * T];

    // ---- Row max (bank 0, write->barrier->broadcast-read) ----
    float m = -3.402823466e+38f;
#pragma unroll
    for (int j = 0; j < CPT; ++j)
      m = fmaxf(m, fmaxf(fmaxf(v[j].x, v[j].y), fmaxf(v[j].z, v[j].w)));
    m = wave_rmax(m);
    if (lane == 0) red_s[0][w] = m;
    __syncthreads();
    m = red_s[0][0];
#pragma unroll
    for (int q = 1; q < NWAVES; ++q) m = fmaxf(m, red_s[0][q]);

    // ---- Newton on f(tau) = sum(max(0, x - tau)) - 1 ----
    // tau0 = m - 1 => f(tau0) >= 0; convex piecewise-linear f => monotone
    // convergence from below; root < m so support count K >= 1 always.
    // tau is block-uniform, so the early-exit branch is uniform + deterministic.
    float tau = m - 1.0f;
#pragma unroll 1
    for (int it = 0; it < NEWTON_MAX; ++it) {
      float s = 0.0f, k = 0.0f;
#pragma unroll
      for (int j = 0; j < CPT; ++j) {
        if (v[j].x > tau) { s += v[j].x; k += 1.0f; }
        if (v[j].y > tau) { s += v[j].y; k += 1.0f; }
        if (v[j].z > tau) { s += v[j].z; k += 1.0f; }
        if (v[j].w > tau) { s += v[j].w; k += 1.0f; }
      }
      s = wave_rsum(s);
      k = wave_rsum(k);
      const int bank = (it + 1) & 1;   // row-max used bank 0; alternate from 1
      if (lane == 0) { red_s[bank][w] = s; red_k[bank][w] = k; }
      __syncthreads();                 // the ONLY barrier in the iteration
      float S = red_s[bank][0], K = red_k[bank][0];
#pragma unroll
      for (int q = 1; q < NWAVES; ++q) { S += red_s[bank][q]; K += red_k[bank][q]; }
      const float tau_new = (S - 1.0f) / K;
      if (tau_new == tau) break;       // uniform: identical in every thread
      tau = tau_new;
    }

    // p = max(0, x - tau), streamed out with non-temporal 128-bit stores.
    v4f* o4 = (v4f*)(out + (size_t)row * D);
#pragma unroll
    for (int j = 0; j < CPT; ++j) {
      v4f r;
      r.x = fmaxf(0.0f, v[j].x - tau);
      r.y = fmaxf(0.0f, v[j].y - tau);
      r.z = fmaxf(0.0f, v[j].z - tau);
      r.w = fmaxf(0.0f, v[j].w - tau);
      __builtin_nontemporal_store(r, &o4[t + j * T]);
    }
    __syncthreads();  // all reads of buf[b] done before next TDM overwrites it
  }
}

extern "C" void kernel_launch(void* const* d_in, const int* in_sizes, int n_in,
                              void* d_out, int out_size, void* d_ws, size_t ws_size,
                              hipStream_t stream) {
  const float* x = (const float*)d_in[0];
  float* out = (float*)d_out;
  const int rows = in_sizes[0] / D;     // 4096 for the reference shapes
  int grid = rows < 1024 ? rows : 1024; // 4 rows per block -> deep TDM pipeline
  if (grid < 1) grid = 1;
  sparsemax_rows_kernel<<<grid, T, 0, stream>>>(x, out, rows);
}